// LightGCN_65506841198665
// MI455X (gfx1250) — compile-verified
//
#include <hip/hip_runtime.h>

#define DIM    64
#define CHUNK  256
#define BLOCK  256

// ---------------- CDNA5 async global->LDS support (probe-guarded) ----------------
#if defined(__AMDGCN__) && __has_builtin(__builtin_amdgcn_global_load_async_to_lds_b32)
#define HAVE_ASYNC_LDS 1
#else
#define HAVE_ASYNC_LDS 0
#endif

#if HAVE_ASYNC_LDS
typedef __attribute__((address_space(1))) int g1_int;
typedef __attribute__((address_space(3))) int l3_int;
__device__ __forceinline__ void async_copy_b32(const void* gsrc, void* ldst) {
  __builtin_amdgcn_global_load_async_to_lds_b32(
      (g1_int*)gsrc, (l3_int*)ldst, /*imm offset=*/0, /*cpol=*/0);
}
#if __has_builtin(__builtin_amdgcn_s_wait_asynccnt)
#define WAIT_ASYNC(n) __builtin_amdgcn_s_wait_asynccnt(n)
#else
#define WAIT_ASYNC(n) asm volatile("s_wait_asynccnt %0" ::"n"(n) : "memory")
#endif
#endif

// Hardware no-return f32 atomic add (global_atomic_add_f32, STOREcnt path).
__device__ __forceinline__ void atomAddF32(float* p, float v) {
#if defined(__HIP_DEVICE_COMPILE__)
  unsafeAtomicAdd(p, v);
#else
  atomicAdd(p, v);
#endif
}

// ---------------- elementwise kernels ----------------
__global__ void lgcn_init(const float4* __restrict__ user, const float4* __restrict__ item,
                          float4* __restrict__ out, float4* __restrict__ prev,
                          float4* __restrict__ cur, int userN4, int totalN4) {
  int i = blockIdx.x * blockDim.x + threadIdx.x;
  if (i >= totalN4) return;
  float4 v = (i < userN4) ? user[i] : item[i - userN4];
  out[i]  = v;
  prev[i] = v;
  cur[i]  = make_float4(0.f, 0.f, 0.f, 0.f);
}

__global__ void lgcn_accum_zero(float4* __restrict__ out, const float4* __restrict__ cur,
                                float4* __restrict__ nextCur, int n4) {
  int i = blockIdx.x * blockDim.x + threadIdx.x;
  if (i >= n4) return;
  float4 o = out[i];
  float4 c = cur[i];
  o.x += c.x; o.y += c.y; o.z += c.z; o.w += c.w;
  out[i] = o;
  nextCur[i] = make_float4(0.f, 0.f, 0.f, 0.f);
}

__global__ void lgcn_accum_scale(float4* __restrict__ out, const float4* __restrict__ cur,
                                 float scale, int n4) {
  int i = blockIdx.x * blockDim.x + threadIdx.x;
  if (i >= n4) return;
  float4 o = out[i];
  float4 c = cur[i];
  o.x = (o.x + c.x) * scale;
  o.y = (o.y + c.y) * scale;
  o.z = (o.z + c.z) * scale;
  o.w = (o.w + c.w) * scale;
  out[i] = o;
}

// ---------------- SpMM: dst[r] += v * src[c], edge-parallel ----------------
// 16 lanes per edge (one float4 slice each). Edge triples double-buffered in LDS
// via async global->LDS copies; s_wait_asynccnt 3 overlaps next-chunk staging
// with current-chunk gather+atomic-scatter.
__global__ void __launch_bounds__(BLOCK) lgcn_spmm(
    const int* __restrict__ rows, const int* __restrict__ cols,
    const float* __restrict__ vals, const float* __restrict__ src,
    float* __restrict__ dst, int nEdges, int nChunks) {
  __shared__ unsigned int sRow[2][CHUNK];
  __shared__ unsigned int sCol[2][CHUNK];
  __shared__ unsigned int sVal[2][CHUNK];

  const int t      = threadIdx.x;
  const int sg     = t >> 4;   // 16 sub-groups of 16 lanes
  const int ln     = t & 15;   // lane within sub-group -> dim slice
  const int stride = gridDim.x;

  int chunk = blockIdx.x;
  if (chunk >= nChunks) return;

  // Stage first chunk. Clamp tail index (instead of predicating) so every wave
  // issues exactly 3 async ops per stage -> ASYNCcnt bookkeeping stays uniform.
  {
    int idx = chunk * CHUNK + t;
    idx = idx < nEdges ? idx : nEdges - 1;
#if HAVE_ASYNC_LDS
    async_copy_b32(rows + idx, &sRow[0][t]);
    async_copy_b32(cols + idx, &sCol[0][t]);
    async_copy_b32(vals + idx, &sVal[0][t]);
#else
    sRow[0][t] = (unsigned)rows[idx];
    sCol[0][t] = (unsigned)cols[idx];
    sVal[0][t] = __float_as_uint(vals[idx]);
#endif
  }

  int buf = 0;
  for (; chunk < nChunks; chunk += stride) {
    const int  nextChunk = chunk + stride;
    const bool hasNext   = nextChunk < nChunks;   // uniform per block

    if (hasNext) {
      int idx = nextChunk * CHUNK + t;
      idx = idx < nEdges ? idx : nEdges - 1;
#if HAVE_ASYNC_LDS
      async_copy_b32(rows + idx, &sRow[buf ^ 1][t]);
      async_copy_b32(cols + idx, &sCol[buf ^ 1][t]);
      async_copy_b32(vals + idx, &sVal[buf ^ 1][t]);
#else
      sRow[buf ^ 1][t] = (unsigned)rows[idx];
      sCol[buf ^ 1][t] = (unsigned)cols[idx];
      sVal[buf ^ 1][t] = __float_as_uint(vals[idx]);
#endif
    }
#if HAVE_ASYNC_LDS
    // async loads complete in order: <=3 outstanding means current buffer landed
    if (hasNext) { WAIT_ASYNC(3); } else { WAIT_ASYNC(0); }
#endif
    __syncthreads();

    const int base = chunk * CHUNK;
    if (base + CHUNK <= nEdges) {
      // full chunk: no per-edge bounds check in the hot loop
#pragma unroll
      for (int p = 0; p < CHUNK / 16; ++p) {
        const int   e = p * 16 + sg;
        const int   r = (int)sRow[buf][e];
        const int   c = (int)sCol[buf][e];
        const float v = __uint_as_float(sVal[buf][e]);
        // coalesced 256B row gather: 16 lanes x b128
        const float4 f = *(const float4*)(src + c * DIM + ln * 4);
        float* dp = dst + r * DIM + ln * 4;
        atomAddF32(dp + 0, v * f.x);
        atomAddF32(dp + 1, v * f.y);
        atomAddF32(dp + 2, v * f.z);
        atomAddF32(dp + 3, v * f.w);
      }
    } else {
      // tail chunk
#pragma unroll
      for (int p = 0; p < CHUNK / 16; ++p) {
        const int e = p * 16 + sg;
        if (base + e < nEdges) {
          const int   r = (int)sRow[buf][e];
          const int   c = (int)sCol[buf][e];
          const float v = __uint_as_float(sVal[buf][e]);
          const float4 f = *(const float4*)(src + c * DIM + ln * 4);
          float* dp = dst + r * DIM + ln * 4;
          atomAddF32(dp + 0, v * f.x);
          atomAddF32(dp + 1, v * f.y);
          atomAddF32(dp + 2, v * f.z);
          atomAddF32(dp + 3, v * f.w);
        }
      }
    }
    __syncthreads();   // everyone done reading buf before it is re-staged
    buf ^= 1;
  }
}

// ---------------- launch ----------------
extern "C" void kernel_launch(void* const* d_in, const int* in_sizes, int n_in,
                              void* d_out, int out_size, void* d_ws, size_t ws_size,
                              hipStream_t stream) {
  const float* user = (const float*)d_in[0];
  const float* item = (const float*)d_in[1];
  const int*   rows = (const int*)d_in[2];
  const int*   cols = (const int*)d_in[3];
  const float* vals = (const float*)d_in[4];
  float*       out  = (float*)d_out;

  const int E       = in_sizes[2];
  const int userN4  = in_sizes[0] / 4;
  const int totalN4 = (in_sizes[0] + in_sizes[1]) / 4;
  const int nFloats = in_sizes[0] + in_sizes[1];

  float* bufA = (float*)d_ws;           // ping
  float* bufB = bufA + nFloats;         // pong  (2 * 38.4 MB of ws)

  dim3 blk(BLOCK);
  const int ewGrid   = (totalN4 + BLOCK - 1) / BLOCK;
  const int nChunks  = (E + CHUNK - 1) / CHUNK;
  const int spmmGrid = (nChunks + 3) / 4;   // ~4 chunks per block -> real double-buffering

  // acc = emb ; prev = emb ; cur(=bufB) = 0
  lgcn_init<<<ewGrid, blk, 0, stream>>>((const float4*)user, (const float4*)item,
                                        (float4*)out, (float4*)bufA, (float4*)bufB,
                                        userN4, totalN4);
  // layer 1: bufB = A @ bufA
  lgcn_spmm<<<spmmGrid, blk, 0, stream>>>(rows, cols, vals, bufA, bufB, E, nChunks);
  // acc += bufB ; zero bufA for next layer
  lgcn_accum_zero<<<ewGrid, blk, 0, stream>>>((float4*)out, (const float4*)bufB,
                                              (float4*)bufA, totalN4);
  // layer 2: bufA = A @ bufB
  lgcn_spmm<<<spmmGrid, blk, 0, stream>>>(rows, cols, vals, bufB, bufA, E, nChunks);
  // acc = (acc + bufA) / 3
  lgcn_accum_scale<<<ewGrid, blk, 0, stream>>>((float4*)out, (const float4*)bufA,
                                               1.0f / 3.0f, totalN4);
}